// mlp_projection_filter_43868795961807
// MI455X (gfx1250) — compile-verified
//
#include <hip/hip_runtime.h>
#include <math.h>

// ---------------------------------------------------------------------------
// CDNA5 (gfx1250) ADMM projection solver, v2.
// All matmuls use V_WMMA_F32_16X16X4_F32 (wave32). One wave per (16-row
// batch tile, channel). Operand staging via LDS using the ISA VGPR layouts:
//   A (16x4 f32):  lane l -> M = l%16, K = {2*(l/16), 2*(l/16)+1}
//   B (4x16 f32):  lane l -> K = {2*(l/16), 2*(l/16)+1}, N = l%16
//   C/D (16x16):   VGPR v, lane l -> M = v + 8*(l/16), N = l%16
// v2: s@A is propagated by the identity s_new@A = rA + g - c@G (G = A^T A),
// eliminating the per-iteration 96-WMMA reconstruction from y_old.
// ---------------------------------------------------------------------------

typedef __attribute__((ext_vector_type(2))) float v2f;
typedef __attribute__((ext_vector_type(8))) float v8f;

#define RELU(x) fmaxf((x), 0.0f)

#define B_BATCH 8192
#define N_ITER  15
#define O_AF    (3 * B_BATCH * 16)          // 393216
#define O_AP    (O_AF + B_BATCH)            // 401408
#define O_PS    (O_AP + B_BATCH)            // 409600
#define O_FS    (O_PS + N_ITER * B_BATCH)   // 532480

// workspace layout (floats)
#define W_MPAD  0            // 20*16
#define W_G     320          // 3*16
#define W_GMAT  368          // 16*16
#define W_RES   640          // 3*15*8192
#define W_FIX   (W_RES + 3 * N_ITER * B_BATCH)

// segB[ch] = {cmax, -cmin, dmax, -dmin, ddmax, -ddmin} per LIM_* constants
static __device__ __constant__ float d_segB[3][6] = {
    {20.0f, -12.0f, 3.00f, 3.00f, 3.00f, 3.00f},   // LIM_V
    {0.20f,  0.20f, 0.25f, 0.25f, 0.15f, 0.15f},   // LIM_PITCH
    {0.25f,  0.25f, 0.25f, 0.25f, 0.15f, 0.15f},   // LIM_ROLL
};

__device__ __forceinline__ v8f wmma4(v2f a, v2f b, v8f c) {
    // D = A(16x4) * B(4x16) + C ; f32 WMMA, 8-arg form
    return __builtin_amdgcn_wmma_f32_16x16x4_f32(false, a, false, b,
                                                 (short)0, c, false, false);
}

// ---------------------------------------------------------------------------
// Setup: cost_mat (18x18) build + Gauss-Jordan inverse; Mpad (20x16) with
// Mpad[k][n] = inv(cost_mat)[n][k] (zero-padded K rows 18..19);
// g[ch][n] = b_comb_ch @ A_control; Gmat = A_control^T A_control (16x16).
// ---------------------------------------------------------------------------
__global__ void admm_setup_kernel(const float* __restrict__ P0,
                                  const float* __restrict__ P1,
                                  const float* __restrict__ P2,
                                  float* __restrict__ Mpad,
                                  float* __restrict__ g,
                                  float* __restrict__ Gmat) {
    __shared__ float cm[18][18];
    __shared__ float aug[18][36];
    const int t = threadIdx.x;  // 32 threads

    for (int e = t; e < 256; e += 32) {
        const int i = e >> 4, j = e & 15;
        float s = 0.0f;
        for (int r = 0; r < 128; ++r)
            s += P0[r * 16 + i] * P0[r * 16 + j] +
                 P1[r * 16 + i] * P1[r * 16 + j] +
                 P2[r * 16 + i] * P2[r * 16 + j];
        Gmat[e] = 2.0f * s;                            // A^T A
        cm[i][j] = 2.0f * s + (i == j ? 1.0f : 0.0f);  // RHO = 1
    }
    if (t < 16) {
        cm[16][t] = P0[t]; cm[17][t] = P1[t];  // A_eq rows
        cm[t][16] = P0[t]; cm[t][17] = P1[t];  // A_eq^T cols
    }
    if (t == 0) { cm[16][16] = 0.f; cm[16][17] = 0.f; cm[17][16] = 0.f; cm[17][17] = 0.f; }
    __syncthreads();

    if (t == 0) {  // Gauss-Jordan with partial pivoting (tiny, one-off)
        for (int i = 0; i < 18; ++i)
            for (int j = 0; j < 36; ++j)
                aug[i][j] = (j < 18) ? cm[i][j] : ((j - 18 == i) ? 1.0f : 0.0f);
        for (int col = 0; col < 18; ++col) {
            int piv = col; float best = fabsf(aug[col][col]);
            for (int r = col + 1; r < 18; ++r) {
                const float v = fabsf(aug[r][col]);
                if (v > best) { best = v; piv = r; }
            }
            if (piv != col)
                for (int j = 0; j < 36; ++j) {
                    const float tmp = aug[col][j]; aug[col][j] = aug[piv][j]; aug[piv][j] = tmp;
                }
            const float dinv = 1.0f / aug[col][col];
            for (int j = 0; j < 36; ++j) aug[col][j] *= dinv;
            for (int r = 0; r < 18; ++r) {
                if (r == col) continue;
                const float f = aug[r][col];
                for (int j = 0; j < 36; ++j) aug[r][j] -= f * aug[col][j];
            }
        }
    }
    __syncthreads();

    for (int e = t; e < 20 * 16; e += 32) {
        const int k = e >> 4, n = e & 15;
        Mpad[e] = (k < 18) ? aug[n][18 + k] : 0.0f;
    }
    if (t < 16) {
        float cs0 = 0.f, cs1 = 0.f, cs2 = 0.f;
        for (int r = 0; r < 128; ++r) {
            cs0 += P0[r * 16 + t]; cs1 += P1[r * 16 + t]; cs2 += P2[r * 16 + t];
        }
        for (int c = 0; c < 3; ++c)
            g[c * 16 + t] = (d_segB[c][0] - d_segB[c][1]) * cs0 +
                            (d_segB[c][2] - d_segB[c][3]) * cs1 +
                            (d_segB[c][4] - d_segB[c][5]) * cs2;
    }
}

// Per-row sum of D-layout partials: row r total = sum over 16 lanes (N dim).
__device__ __forceinline__ float rowreduce16(const v8f acc, float (*red)[17],
                                             int h, int m16) {
    __syncthreads();
#pragma unroll
    for (int v = 0; v < 8; ++v) red[v + 8 * h][m16] = acc[v];
    __syncthreads();
    float s = 0.0f;
#pragma unroll
    for (int k = 0; k < 16; ++k) s += red[m16][k];
    return s;  // lane's m16 indexes the row; both halves agree
}

// ---------------------------------------------------------------------------
// Main solver: one wave32 per (16-row tile, channel). 15 iterations.
// ---------------------------------------------------------------------------
__global__ __launch_bounds__(32) void admm_solve_kernel(
    const float* __restrict__ P0, const float* __restrict__ P1, const float* __restrict__ P2,
    const float* __restrict__ lam_v, const float* __restrict__ lam_p, const float* __restrict__ lam_r,
    const float* __restrict__ ci_v, const float* __restrict__ ci_p, const float* __restrict__ ci_r,
    const float* __restrict__ be_v, const float* __restrict__ be_p, const float* __restrict__ be_r,
    const float* __restrict__ c0_v, const float* __restrict__ c0_p, const float* __restrict__ c0_r,
    const float* __restrict__ Mpad_g, const float* __restrict__ g_g, const float* __restrict__ Gmat_g,
    float* __restrict__ res_ws, float* __restrict__ fix_ws, float* __restrict__ out) {
    __shared__ float s_pm[3][128][16];  // P, Pdot, Pddot row-major
    __shared__ float s_y[16][388];      // y state per tile row (stride 388: bank-safe)
    __shared__ float s_M[20][16];       // padded inverse, B-matrix order
    __shared__ float s_G[16][16];       // A^T A, B-matrix order
    __shared__ float s_stage[16][20];   // D-layout -> A-layout bounce buffer
    __shared__ float s_red[16][17];     // row-reduction scratch

    const int lane = threadIdx.x;
    const int h = lane >> 4;
    const int m16 = lane & 15;
    const int ch = blockIdx.x % 3;
    const int tile = blockIdx.x / 3;
    const int rowBase = tile * 16;

    const float* lam0 = (ch == 0) ? lam_v : (ch == 1) ? lam_p : lam_r;
    const float* ci   = (ch == 0) ? ci_v  : (ch == 1) ? ci_p  : ci_r;
    const float* be   = (ch == 0) ? be_v  : (ch == 1) ? be_p  : be_r;
    const float* c0   = (ch == 0) ? c0_v  : (ch == 1) ? c0_p  : c0_r;

    // Prefetch this tile's state rows (global_prefetch_b8 path).
    __builtin_prefetch(&c0[rowBase * 16], 0, 3);
    __builtin_prefetch(&lam0[rowBase * 16], 0, 3);
    __builtin_prefetch(&ci[rowBase * 16], 0, 3);

    for (int i = lane; i < 128 * 16; i += 32) {
        s_pm[0][0][i] = P0[i];
        s_pm[1][0][i] = P1[i];
        s_pm[2][0][i] = P2[i];
    }
    for (int i = lane; i < 20 * 16; i += 32) (&s_M[0][0])[i] = Mpad_g[i];
    for (int i = lane; i < 16 * 16; i += 32) (&s_G[0][0])[i] = Gmat_g[i];
    __syncthreads();

    v8f zero8;
#pragma unroll
    for (int v = 0; v < 8; ++v) zero8[v] = 0.0f;

    // Per-wave state in C/D layout.
    v8f c_d, lam_d, ci_d;
#pragma unroll
    for (int v = 0; v < 8; ++v) {
        const int r = rowBase + v + 8 * h;
        c_d[v]   = c0[r * 16 + m16];
        lam_d[v] = lam0[r * 16 + m16];
        ci_d[v]  = ci[r * 16 + m16];
    }
    const float gval = g_g[ch * 16 + m16];
    float beq0 = 0.0f, beq1 = 0.0f;
    if (h == 0) {
        beq0 = be[(rowBase + m16) * 2 + 0];
        beq1 = be[(rowBase + m16) * 2 + 1];
    }

    // ---- init: y0 = c0 @ Pmat_j^T (defines s0 via relu) ----
#pragma unroll
    for (int v = 0; v < 8; ++v) s_stage[v + 8 * h][m16] = c_d[v];
    __syncthreads();
    for (int j = 0; j < 3; ++j)
        for (int n = 0; n < 8; ++n) {
            v8f y = zero8;
#pragma unroll
            for (int kk = 0; kk < 4; ++kk) {
                v2f a, b;
                a.x = s_stage[m16][4 * kk + 2 * h];
                a.y = s_stage[m16][4 * kk + 2 * h + 1];
                const int pr = n * 16 + m16;
                b.x = s_pm[j][pr][4 * kk + 2 * h];      // B[k][p] = Pmat[p][k]
                b.y = s_pm[j][pr][4 * kk + 2 * h + 1];
                y = wmma4(a, b, y);
            }
            const int col = j * 128 + n * 16 + m16;
#pragma unroll
            for (int v = 0; v < 8; ++v) s_y[v + 8 * h][col] = y[v];
        }
    __syncthreads();

    // ---- one-time: sA0 = s0 @ A_control from y0 relus ----
    v8f sA = zero8;
    for (int j = 0; j < 3; ++j) {
        const float b0 = d_segB[ch][2 * j], b1 = d_segB[ch][2 * j + 1];
        for (int n = 0; n < 8; ++n) {
#pragma unroll
            for (int kk = 0; kk < 4; ++kk) {
                const int colb = j * 128 + n * 16 + 4 * kk + 2 * h;
                const float ya = s_y[m16][colb];
                const float yb = s_y[m16][colb + 1];
                v2f a, b;
                a.x = RELU(b0 - ya) - RELU(b1 + ya);
                a.y = RELU(b0 - yb) - RELU(b1 + yb);
                const int kr = n * 16 + 4 * kk + 2 * h;
                b.x = s_pm[j][kr][m16];
                b.y = s_pm[j][kr + 1][m16];
                sA = wmma4(a, b, sA);
            }
        }
    }

    // ---- 15 ADMM iterations ----
    for (int it = 0; it < N_ITER; ++it) {
        v8f accR = zero8, accS = zero8;

        // Phase B: rhs = [lam + ci + g - sA, b_eq, 0, 0]; c_new = rhs @ Minv^T
        __syncthreads();
#pragma unroll
        for (int v = 0; v < 8; ++v)
            s_stage[v + 8 * h][m16] = lam_d[v] + ci_d[v] + gval - sA[v];
        if (h == 0) {
            s_stage[m16][16] = beq0; s_stage[m16][17] = beq1;
            s_stage[m16][18] = 0.0f; s_stage[m16][19] = 0.0f;
        }
        __syncthreads();
        v8f cn = zero8;
#pragma unroll
        for (int kk = 0; kk < 5; ++kk) {  // K = 20 (18 padded)
            v2f a, b;
            a.x = s_stage[m16][4 * kk + 2 * h];
            a.y = s_stage[m16][4 * kk + 2 * h + 1];
            b.x = s_M[4 * kk + 2 * h][m16];
            b.y = s_M[4 * kk + 2 * h + 1][m16];
            cn = wmma4(a, b, cn);
        }
        v8f accC = zero8;
#pragma unroll
        for (int v = 0; v < 8; ++v) {
            const float d = cn[v] - c_d[v];
            accC[v] = d * d;
            c_d[v] = cn[v];
        }
        __syncthreads();
#pragma unroll
        for (int v = 0; v < 8; ++v) s_stage[v + 8 * h][m16] = c_d[v];
        __syncthreads();

        // cG = c_new @ G (for the sA recurrence), from the staged c_new
        v8f cG = zero8;
#pragma unroll
        for (int kk = 0; kk < 4; ++kk) {
            v2f a, b;
            a.x = s_stage[m16][4 * kk + 2 * h];
            a.y = s_stage[m16][4 * kk + 2 * h + 1];
            b.x = s_G[4 * kk + 2 * h][m16];
            b.y = s_G[4 * kk + 2 * h + 1][m16];
            cG = wmma4(a, b, cG);
        }

        // Phase C/D per 16x16 tile: y_new = c @ Pmat^T, residual relus,
        // ||s_new - s|| vs y_old, then rA += relu-diff(y_new) @ Pmat
        v8f rA = zero8;
        for (int j = 0; j < 3; ++j) {
            const float b0 = d_segB[ch][2 * j], b1 = d_segB[ch][2 * j + 1];
            for (int n = 0; n < 8; ++n) {
                v8f y = zero8;
#pragma unroll
                for (int kk = 0; kk < 4; ++kk) {
                    v2f a, b;
                    a.x = s_stage[m16][4 * kk + 2 * h];
                    a.y = s_stage[m16][4 * kk + 2 * h + 1];
                    const int pr = n * 16 + m16;
                    b.x = s_pm[j][pr][4 * kk + 2 * h];
                    b.y = s_pm[j][pr][4 * kk + 2 * h + 1];
                    y = wmma4(a, b, y);
                }
                const int col = j * 128 + n * 16 + m16;
#pragma unroll
                for (int v = 0; v < 8; ++v) {
                    const float yo = s_y[v + 8 * h][col];
                    const float yn = y[v];
                    const float rp = RELU(yn - b0), rm = RELU(-yn - b1);
                    accR[v] += rp * rp + rm * rm;
                    const float d1 = RELU(b0 - yn) - RELU(b0 - yo);
                    const float d2 = RELU(b1 + yn) - RELU(b1 + yo);
                    accS[v] += d1 * d1 + d2 * d2;
                    s_y[v + 8 * h][col] = yn;  // y_old <- y_new
                }
                __syncthreads();
#pragma unroll
                for (int kk = 0; kk < 4; ++kk) {
                    const int colb = j * 128 + n * 16 + 4 * kk + 2 * h;
                    const float ya = s_y[m16][colb];
                    const float yb = s_y[m16][colb + 1];
                    v2f a, b;
                    a.x = RELU(ya - b0) - RELU(-ya - b1);
                    a.y = RELU(yb - b0) - RELU(-yb - b1);
                    const int kr = n * 16 + 4 * kk + 2 * h;
                    b.x = s_pm[j][kr][m16];
                    b.y = s_pm[j][kr + 1][m16];
                    rA = wmma4(a, b, rA);
                }
            }
        }

        // lamda_new = lamda - res_vec @ A_control (RHO=1)
        v8f accL = zero8;
#pragma unroll
        for (int v = 0; v < 8; ++v) {
            accL[v] = rA[v] * rA[v];
            lam_d[v] -= rA[v];
        }

        // sA recurrence: s_new@A = rA + g - c@G
#pragma unroll
        for (int v = 0; v < 8; ++v) sA[v] = rA[v] + gval - cG[v];

        // Per-row norms -> workspace (channel-separated, deterministic)
        const float sR = rowreduce16(accR, s_red, h, m16);
        const float sC = rowreduce16(accC, s_red, h, m16);
        const float sL = rowreduce16(accL, s_red, h, m16);
        const float sS = rowreduce16(accS, s_red, h, m16);
        if (h == 0) {
            const int idx = (ch * N_ITER + it) * B_BATCH + rowBase + m16;
            res_ws[idx] = sqrtf(sR);
            fix_ws[idx] = sqrtf(sC) + sqrtf(sL) + sqrtf(sS);
        }
    }

    // Final c for this channel
    float* oc = out + ch * (B_BATCH * 16);
#pragma unroll
    for (int v = 0; v < 8; ++v)
        oc[(rowBase + v + 8 * h) * 16 + m16] = c_d[v];
}

// ---------------------------------------------------------------------------
// Finalize: sum channels into primal/fixed stacks and compute means.
// ---------------------------------------------------------------------------
__global__ void admm_finalize_kernel(const float* __restrict__ res_ws,
                                     const float* __restrict__ fix_ws,
                                     float* __restrict__ out) {
    const int r = blockIdx.x * blockDim.x + threadIdx.x;
    if (r >= B_BATCH) return;
    float ap = 0.0f, af = 0.0f;
    for (int it = 0; it < N_ITER; ++it) {
        float p = 0.0f, f = 0.0f;
        for (int c = 0; c < 3; ++c) {
            p += res_ws[(c * N_ITER + it) * B_BATCH + r];
            f += fix_ws[(c * N_ITER + it) * B_BATCH + r];
        }
        out[O_PS + it * B_BATCH + r] = p;
        out[O_FS + it * B_BATCH + r] = f;
        ap += p; af += f;
    }
    out[O_AF + r] = af * (1.0f / 15.0f);
    out[O_AP + r] = ap * (1.0f / 15.0f);
}

extern "C" void kernel_launch(void* const* d_in, const int* in_sizes, int n_in,
                              void* d_out, int out_size, void* d_ws, size_t ws_size,
                              hipStream_t stream) {
    (void)in_sizes; (void)n_in; (void)out_size; (void)ws_size;
    const float* P0   = (const float*)d_in[0];
    const float* P1   = (const float*)d_in[1];
    const float* P2   = (const float*)d_in[2];
    const float* lamv = (const float*)d_in[3];
    const float* lamp = (const float*)d_in[4];
    const float* lamr = (const float*)d_in[5];
    const float* civ  = (const float*)d_in[6];
    const float* cip  = (const float*)d_in[7];
    const float* cir  = (const float*)d_in[8];
    const float* bev  = (const float*)d_in[9];
    const float* bep  = (const float*)d_in[10];
    const float* ber  = (const float*)d_in[11];
    const float* c0v  = (const float*)d_in[12];
    const float* c0p  = (const float*)d_in[13];
    const float* c0r  = (const float*)d_in[14];

    float* ws     = (float*)d_ws;
    float* Mpad   = ws + W_MPAD;
    float* g      = ws + W_G;
    float* Gmat   = ws + W_GMAT;
    float* res_ws = ws + W_RES;
    float* fix_ws = ws + W_FIX;
    float* out    = (float*)d_out;

    admm_setup_kernel<<<1, 32, 0, stream>>>(P0, P1, P2, Mpad, g, Gmat);
    admm_solve_kernel<<<(B_BATCH / 16) * 3, 32, 0, stream>>>(
        P0, P1, P2, lamv, lamp, lamr, civ, cip, cir, bev, bep, ber,
        c0v, c0p, c0r, Mpad, g, Gmat, res_ws, fix_ws, out);
    admm_finalize_kernel<<<(B_BATCH + 255) / 256, 256, 0, stream>>>(res_ws, fix_ws, out);
}